// GAPNet_readout_embedding_32006096289957
// MI455X (gfx1250) — compile-verified
//
#include <hip/hip_runtime.h>
#include <hip/hip_bf16.h>
#include <math.h>

// ---------------- problem constants ----------------
constexpr int Bg   = 64;     // graphs
constexpr int Nn   = 512;    // nodes per graph
constexpr int FIN  = 128;
constexpr int Hd   = 64;
constexpr int OUTd = 10;
constexpr int K2d  = 16;
constexpr int Etot = Bg * Nn * 16;
constexpr float MUH = 0.05f; // MU * 0.5 (symmetrized g = 0.5*(fu-fv)^2)
constexpr float EPSf = 1e-15f;

typedef __attribute__((ext_vector_type(2))) float v2f;
typedef __attribute__((ext_vector_type(8))) float v8f;

static __device__ __forceinline__ v8f wmma4(v2f a, v2f b, v8f c) {
  // V_WMMA_F32_16X16X4_F32 : D = A(16x4) * B(4x16) + C(16x16), all fp32
  return __builtin_amdgcn_wmma_f32_16x16x4_f32(false, a, false, b, (short)0, c,
                                               false, false);
}

// ---------------- adj build ----------------
__global__ void k_build_adj(const int* __restrict__ ei, float* __restrict__ adj) {
  int e = blockIdx.x * blockDim.x + threadIdx.x;
  if (e >= Etot) return;
  int src = ei[e];
  int dst = ei[Etot + e];
  int g = src >> 9;             // src / Nn (batch is sorted, N per graph)
  int u = src & (Nn - 1);
  int v = dst & (Nn - 1);
  atomicAdd(&adj[(size_t)g * Nn * Nn + (size_t)u * Nn + v], 1.0f);
}

// ---------------- lin1: h = x @ W1 + b1   (M=32768,K=128,N=64) ----------------
__global__ void k_lin1(const float* __restrict__ x, const float* __restrict__ w,
                       const float* __restrict__ bias, float* __restrict__ h) {
  const int wave = threadIdx.x >> 5, lane = threadIdx.x & 31;
  const int tile = blockIdx.x * 4 + wave;       // 2048 row-tiles of 16
  const int row0 = tile * 16;
  const int mrow = row0 + (lane & 15);
  const int kb   = (lane >> 4) * 2;             // lanes 0-15: K{0,1}; 16-31: K{2,3}
  const int cl   = lane & 15;
  v8f a0 = {}, a1 = {}, a2 = {}, a3 = {};
  for (int k0 = 0; k0 < FIN; k0 += 4) {
    v2f a = *(const v2f*)(x + (size_t)mrow * FIN + k0 + kb);
    const float* w0 = w + (size_t)(k0 + kb) * Hd;
    const float* w1 = w0 + Hd;
    v2f b0, b1, b2, b3;
    b0.x = w0[cl];      b0.y = w1[cl];
    b1.x = w0[16 + cl]; b1.y = w1[16 + cl];
    b2.x = w0[32 + cl]; b2.y = w1[32 + cl];
    b3.x = w0[48 + cl]; b3.y = w1[48 + cl];
    a0 = wmma4(a, b0, a0); a1 = wmma4(a, b1, a1);
    a2 = wmma4(a, b2, a2); a3 = wmma4(a, b3, a3);
  }
  const int rbase = row0 + ((lane >> 4) << 3);
#pragma unroll
  for (int r = 0; r < 8; ++r) {
    size_t ro = (size_t)(rbase + r) * Hd;
    h[ro + cl]      = a0[r] + bias[cl];
    h[ro + 16 + cl] = a1[r] + bias[16 + cl];
    h[ro + 32 + cl] = a2[r] + bias[32 + cl];
    h[ro + 48 + cl] = a3[r] + bias[48 + cl];
  }
}

// ---------------- pool1 logits + 2-way softmax ----------------
__global__ void k_pool1(const float* __restrict__ h, const float* __restrict__ w,
                        const float* __restrict__ b, float* __restrict__ s1) {
  int idx = blockIdx.x * blockDim.x + threadIdx.x;
  if (idx >= Bg * Nn) return;
  const float* hr = h + (size_t)idx * Hd;
  float l0 = b[0], l1 = b[1];
  for (int k = 0; k < Hd; ++k) { float v = hr[k]; l0 += v * w[k * 2]; l1 += v * w[k * 2 + 1]; }
  float m = fmaxf(l0, l1);
  float e0 = expf(l0 - m), e1 = expf(l1 - m);
  float inv = 1.0f / (e0 + e1);
  s1[idx * 2] = e0 * inv; s1[idx * 2 + 1] = e1 * inv;
}

// ---------------- As1 = A @ s1 (k=2) and degree ----------------
__global__ void k_as1_deg(const float* __restrict__ adj, const float* __restrict__ s1,
                          float* __restrict__ As1, float* __restrict__ deg) {
  int idx = blockIdx.x * blockDim.x + threadIdx.x;
  if (idx >= Bg * Nn) return;
  int g = idx >> 9;
  const float* ar = adj + (size_t)g * Nn * Nn + (size_t)(idx & (Nn - 1)) * Nn;
  const float* sb = s1 + (size_t)g * Nn * 2;
  float dg = 0.f, a0 = 0.f, a1 = 0.f;
  for (int m = 0; m < Nn; ++m) {
    float a = ar[m];
    dg += a; a0 += a * sb[m * 2]; a1 += a * sb[m * 2 + 1];
  }
  deg[idx] = dg; As1[idx * 2] = a0; As1[idx * 2 + 1] = a1;
}

// ---------------- stage-1 stats: mincut1, ortho1, Fiedler coefficients ----------------
__global__ void k_stats1(const float* __restrict__ s1, const float* __restrict__ As1,
                         const float* __restrict__ deg, float* __restrict__ coef,
                         float* __restrict__ stat) {
  __shared__ float sh[7][256];
  int g = blockIdx.x, t = threadIdx.x;
  float num = 0, den = 0, ss00 = 0, ss01 = 0, ss11 = 0, n1 = 0, n2 = 0;
  for (int n = t; n < Nn; n += 256) {
    size_t idx = (size_t)g * Nn + n;
    float s0 = s1[idx * 2], sv = s1[idx * 2 + 1];
    float a0 = As1[idx * 2], a1 = As1[idx * 2 + 1];
    float d = deg[idx];
    num += s0 * a0 + sv * a1;
    den += d * (s0 * s0 + sv * sv);
    ss00 += s0 * s0; ss01 += s0 * sv; ss11 += sv * sv;
    n1 += s0; n2 += sv;
  }
  sh[0][t]=num; sh[1][t]=den; sh[2][t]=ss00; sh[3][t]=ss01; sh[4][t]=ss11; sh[5][t]=n1; sh[6][t]=n2;
  __syncthreads();
  for (int s = 128; s > 0; s >>= 1) {
    if (t < s) for (int j = 0; j < 7; ++j) sh[j][t] += sh[j][t + s];
    __syncthreads();
  }
  if (t == 0) {
    num = sh[0][0]; den = sh[1][0]; ss00 = sh[2][0]; ss01 = sh[3][0]; ss11 = sh[4][0];
    n1 = sh[5][0]; n2 = sh[6][0];
    stat[g] = -(num / den);                                   // mincut contribution
    float nrm = sqrtf(ss00 * ss00 + 2.f * ss01 * ss01 + ss11 * ss11);
    float r = 0.70710678f;                                    // 1/sqrt(2)
    float d00 = ss00 / nrm - r, d01 = ss01 / nrm, d11 = ss11 / nrm - r;
    stat[Bg + g] = sqrtf(d00 * d00 + 2.f * d01 * d01 + d11 * d11);  // ortho
    float nT = n1 + n2;
    coef[g * 2]     = sqrtf(n2 / (nT * n1 + EPSf));
    coef[g * 2 + 1] = sqrtf(n1 / (nT * n2 + EPSf));
  }
}

// ---------------- f vector ----------------
__global__ void k_fvec(const float* __restrict__ s1, const float* __restrict__ coef,
                       float* __restrict__ f) {
  int idx = blockIdx.x * blockDim.x + threadIdx.x;
  if (idx >= Bg * Nn) return;
  int g = idx >> 9;
  f[idx] = s1[idx * 2] * coef[g * 2] - s1[idx * 2 + 1] * coef[g * 2 + 1];
}

// ---------------- rewire: A = relu(A - MU*0.5*(fu-fv)^2), float4 ----------------
__global__ void k_rewire(float* __restrict__ adj, const float* __restrict__ f) {
  size_t t = (size_t)blockIdx.x * blockDim.x + threadIdx.x;
  if (t >= (size_t)Bg * Nn * Nn / 4) return;
  size_t base = t * 4;
  int g  = (int)(base >> 18);
  int u  = (int)((base >> 9) & (Nn - 1));
  int v0 = (int)(base & (Nn - 1));
  float fu = f[g * Nn + u];
  const float* fb = f + g * Nn + v0;
  float4 a = *(const float4*)(adj + base);
  float d;
  d = fu - fb[0]; a.x = fmaxf(a.x - MUH * d * d, 0.f);
  d = fu - fb[1]; a.y = fmaxf(a.y - MUH * d * d, 0.f);
  d = fu - fb[2]; a.z = fmaxf(a.z - MUH * d * d, 0.f);
  d = fu - fb[3]; a.w = fmaxf(a.w - MUH * d * d, 0.f);
  *(float4*)(adj + base) = a;
}

// ---------------- Ah = adj @ h  (per-graph 512x512x64, WMMA) ----------------
__global__ void k_spmm(const float* __restrict__ adj, const float* __restrict__ h,
                       float* __restrict__ out) {
  const int wave = threadIdx.x >> 5, lane = threadIdx.x & 31;
  const int tile = blockIdx.x * 4 + wave;        // 2048 tiles: (g, mt)
  const int g = tile >> 5, mt = tile & 31;
  const float* A  = adj + (size_t)g * Nn * Nn;
  const float* Hm = h + (size_t)g * Nn * Hd;
  const int mrow = mt * 16 + (lane & 15);
  const int kb = (lane >> 4) * 2;
  const int cl = lane & 15;
  v8f a0 = {}, a1 = {}, a2 = {}, a3 = {};
  for (int k0 = 0; k0 < Nn; k0 += 4) {
    v2f a = *(const v2f*)(A + (size_t)mrow * Nn + k0 + kb);
    const float* h0 = Hm + (size_t)(k0 + kb) * Hd;
    const float* h1 = h0 + Hd;
    v2f b0, b1, b2, b3;
    b0.x = h0[cl];      b0.y = h1[cl];
    b1.x = h0[16 + cl]; b1.y = h1[16 + cl];
    b2.x = h0[32 + cl]; b2.y = h1[32 + cl];
    b3.x = h0[48 + cl]; b3.y = h1[48 + cl];
    a0 = wmma4(a, b0, a0); a1 = wmma4(a, b1, a1);
    a2 = wmma4(a, b2, a2); a3 = wmma4(a, b3, a3);
  }
  const int rbase = g * Nn + mt * 16 + ((lane >> 4) << 3);
#pragma unroll
  for (int r = 0; r < 8; ++r) {
    size_t ro = (size_t)(rbase + r) * Hd;
    out[ro + cl] = a0[r]; out[ro + 16 + cl] = a1[r];
    out[ro + 32 + cl] = a2[r]; out[ro + 48 + cl] = a3[r];
  }
}

// ---------------- hn = Ah @ relW + h @ rootW + relB  (M=32768,K=64x2,N=64) --------
__global__ void k_convmix(const float* __restrict__ Ah, const float* __restrict__ h,
                          const float* __restrict__ relw, const float* __restrict__ relb,
                          const float* __restrict__ rootw, float* __restrict__ hn) {
  const int wave = threadIdx.x >> 5, lane = threadIdx.x & 31;
  const int tile = blockIdx.x * 4 + wave;
  const int row0 = tile * 16;
  const int mrow = row0 + (lane & 15);
  const int kb = (lane >> 4) * 2;
  const int cl = lane & 15;
  v8f a0 = {}, a1 = {}, a2 = {}, a3 = {};
  for (int pass = 0; pass < 2; ++pass) {
    const float* X = pass ? h : Ah;
    const float* W = pass ? rootw : relw;
    for (int k0 = 0; k0 < Hd; k0 += 4) {
      v2f a = *(const v2f*)(X + (size_t)mrow * Hd + k0 + kb);
      const float* w0 = W + (size_t)(k0 + kb) * Hd;
      const float* w1 = w0 + Hd;
      v2f b0, b1, b2, b3;
      b0.x = w0[cl];      b0.y = w1[cl];
      b1.x = w0[16 + cl]; b1.y = w1[16 + cl];
      b2.x = w0[32 + cl]; b2.y = w1[32 + cl];
      b3.x = w0[48 + cl]; b3.y = w1[48 + cl];
      a0 = wmma4(a, b0, a0); a1 = wmma4(a, b1, a1);
      a2 = wmma4(a, b2, a2); a3 = wmma4(a, b3, a3);
    }
  }
  const int rbase = row0 + ((lane >> 4) << 3);
#pragma unroll
  for (int r = 0; r < 8; ++r) {
    size_t ro = (size_t)(rbase + r) * Hd;
    hn[ro + cl]      = a0[r] + relb[cl];
    hn[ro + 16 + cl] = a1[r] + relb[16 + cl];
    hn[ro + 32 + cl] = a2[r] + relb[32 + cl];
    hn[ro + 48 + cl] = a3[r] + relb[48 + cl];
  }
}

// ---------------- pool2 logits + 16-way softmax ----------------
__global__ void k_pool2(const float* __restrict__ h, const float* __restrict__ w,
                        const float* __restrict__ b, float* __restrict__ s2) {
  int idx = blockIdx.x * blockDim.x + threadIdx.x;
  if (idx >= Bg * Nn) return;
  const float* hr = h + (size_t)idx * Hd;
  float l[K2d];
  for (int j = 0; j < K2d; ++j) l[j] = b[j];
  for (int k = 0; k < Hd; ++k) {
    float v = hr[k];
    const float* wr = w + k * K2d;
    for (int j = 0; j < K2d; ++j) l[j] += v * wr[j];
  }
  float m = l[0];
  for (int j = 1; j < K2d; ++j) m = fmaxf(m, l[j]);
  float se = 0.f;
  for (int j = 0; j < K2d; ++j) { l[j] = expf(l[j] - m); se += l[j]; }
  float inv = 1.f / se;
  float* sr = s2 + (size_t)idx * K2d;
  for (int j = 0; j < K2d; ++j) sr[j] = l[j] * inv;
}

// ---------------- As2 = adj @ s2  (per-graph 512x512x16, WMMA) ----------------
__global__ void k_spmm_s2(const float* __restrict__ adj, const float* __restrict__ s2,
                          float* __restrict__ As2) {
  const int wave = threadIdx.x >> 5, lane = threadIdx.x & 31;
  const int tile = blockIdx.x * 4 + wave;
  const int g = tile >> 5, mt = tile & 31;
  const float* A = adj + (size_t)g * Nn * Nn;
  const float* S = s2 + (size_t)g * Nn * K2d;
  const int mrow = mt * 16 + (lane & 15);
  const int kb = (lane >> 4) * 2;
  const int cl = lane & 15;
  v8f acc = {};
  for (int k0 = 0; k0 < Nn; k0 += 4) {
    v2f a = *(const v2f*)(A + (size_t)mrow * Nn + k0 + kb);
    v2f b;
    b.x = S[(size_t)(k0 + kb) * K2d + cl];
    b.y = S[(size_t)(k0 + kb + 1) * K2d + cl];
    acc = wmma4(a, b, acc);
  }
  const int rbase = g * Nn + mt * 16 + ((lane >> 4) << 3);
#pragma unroll
  for (int r = 0; r < 8; ++r) As2[(size_t)(rbase + r) * K2d + cl] = acc[r];
}

// ---------------- deg2 ----------------
__global__ void k_deg2(const float* __restrict__ adj, float* __restrict__ deg2) {
  int idx = blockIdx.x * blockDim.x + threadIdx.x;
  if (idx >= Bg * Nn) return;
  int g = idx >> 9;
  const float4* ar = (const float4*)(adj + (size_t)g * Nn * Nn + (size_t)(idx & (Nn - 1)) * Nn);
  float dg = 0.f;
  for (int m = 0; m < Nn / 4; ++m) { float4 a = ar[m]; dg += a.x + a.y + a.z + a.w; }
  deg2[idx] = dg;
}

// ---------------- xp = s2^T @ h   (per-graph 16x512x64, WMMA, 1 wave/graph) ------
__global__ void k_poolx(const float* __restrict__ s2, const float* __restrict__ h,
                        float* __restrict__ xp) {
  const int wave = threadIdx.x >> 5, lane = threadIdx.x & 31;
  const int g = blockIdx.x * 4 + wave;
  const float* S  = s2 + (size_t)g * Nn * K2d;
  const float* Hm = h + (size_t)g * Nn * Hd;
  const int am = lane & 15;
  const int kb = (lane >> 4) * 2;
  const int cl = lane & 15;
  v8f a0 = {}, a1 = {}, a2 = {}, a3 = {};
  for (int k0 = 0; k0 < Nn; k0 += 4) {
    v2f a;                                       // A[m][k] = s2[k][m]
    a.x = S[(size_t)(k0 + kb) * K2d + am];
    a.y = S[(size_t)(k0 + kb + 1) * K2d + am];
    const float* h0 = Hm + (size_t)(k0 + kb) * Hd;
    const float* h1 = h0 + Hd;
    v2f b0, b1, b2, b3;
    b0.x = h0[cl];      b0.y = h1[cl];
    b1.x = h0[16 + cl]; b1.y = h1[16 + cl];
    b2.x = h0[32 + cl]; b2.y = h1[32 + cl];
    b3.x = h0[48 + cl]; b3.y = h1[48 + cl];
    a0 = wmma4(a, b0, a0); a1 = wmma4(a, b1, a1);
    a2 = wmma4(a, b2, a2); a3 = wmma4(a, b3, a3);
  }
  const int rbase = ((lane >> 4) << 3);
#pragma unroll
  for (int r = 0; r < 8; ++r) {
    size_t ro = ((size_t)g * K2d + rbase + r) * Hd;
    xp[ro + cl] = a0[r]; xp[ro + 16 + cl] = a1[r];
    xp[ro + 32 + cl] = a2[r]; xp[ro + 48 + cl] = a3[r];
  }
}

// ------- oadj = s2^T @ As2 ; ss = s2^T @ s2  (per-graph 16x512x16, fused WMMA) ----
__global__ void k_oadj_ss(const float* __restrict__ s2, const float* __restrict__ As2,
                          float* __restrict__ oadj, float* __restrict__ ss) {
  const int wave = threadIdx.x >> 5, lane = threadIdx.x & 31;
  const int g = blockIdx.x * 4 + wave;
  const float* S = s2 + (size_t)g * Nn * K2d;
  const float* T = As2 + (size_t)g * Nn * K2d;
  const int am = lane & 15;
  const int kb = (lane >> 4) * 2;
  const int cl = lane & 15;
  v8f ao = {}, as = {};
  for (int k0 = 0; k0 < Nn; k0 += 4) {
    v2f a;
    a.x = S[(size_t)(k0 + kb) * K2d + am];
    a.y = S[(size_t)(k0 + kb + 1) * K2d + am];
    v2f b1, b2;
    b1.x = T[(size_t)(k0 + kb) * K2d + cl];
    b1.y = T[(size_t)(k0 + kb + 1) * K2d + cl];
    b2.x = S[(size_t)(k0 + kb) * K2d + cl];
    b2.y = S[(size_t)(k0 + kb + 1) * K2d + cl];
    ao = wmma4(a, b1, ao);
    as = wmma4(a, b2, as);
  }
  const int rbase = ((lane >> 4) << 3);
#pragma unroll
  for (int r = 0; r < 8; ++r) {
    size_t ro = ((size_t)g * K2d + rbase + r) * K2d + cl;
    oadj[ro] = ao[r];
    ss[ro]   = as[r];
  }
}

// ---------------- stage-2 stats + normalized pooled adjacency ----------------
__global__ void k_stats2(const float* __restrict__ s2, const float* __restrict__ deg2,
                         const float* __restrict__ oadj, const float* __restrict__ ss,
                         float* __restrict__ adj2, float* __restrict__ stat) {
  __shared__ float red[256];
  __shared__ float so[16][16];
  __shared__ float dcol[16];
  __shared__ float sden, snum, sfro;
  int g = blockIdx.x, t = threadIdx.x;
  // den2 = sum_n deg2[n] * sum_k s2[n,k]^2
  float den = 0.f;
  for (int n = t; n < Nn; n += 256) {
    const float* sr = s2 + ((size_t)g * Nn + n) * K2d;
    float acc = 0.f;
    for (int k = 0; k < K2d; ++k) acc += sr[k] * sr[k];
    den += deg2[g * Nn + n] * acc;
  }
  red[t] = den; __syncthreads();
  for (int s = 128; s > 0; s >>= 1) { if (t < s) red[t] += red[t + s]; __syncthreads(); }
  if (t == 0) sden = red[0];
  __syncthreads();
  int i = t >> 4, j = t & 15;
  float ov = oadj[((size_t)g * K2d + i) * K2d + j];
  float sv = ss[((size_t)g * K2d + i) * K2d + j];
  red[t] = (i == j) ? ov : 0.f; __syncthreads();
  for (int s = 128; s > 0; s >>= 1) { if (t < s) red[t] += red[t + s]; __syncthreads(); }
  if (t == 0) snum = red[0];
  __syncthreads();
  red[t] = sv * sv; __syncthreads();
  for (int s = 128; s > 0; s >>= 1) { if (t < s) red[t] += red[t + s]; __syncthreads(); }
  if (t == 0) sfro = sqrtf(red[0]);
  __syncthreads();
  float dd = sv / sfro - ((i == j) ? 0.25f : 0.f);   // eye/sqrt(16)
  red[t] = dd * dd; __syncthreads();
  for (int s = 128; s > 0; s >>= 1) { if (t < s) red[t] += red[t + s]; __syncthreads(); }
  if (t == 0) {
    stat[2 * Bg + g] = -(snum / sden);     // mincut2
    stat[3 * Bg + g] = sqrtf(red[0]);      // ortho2
  }
  __syncthreads();
  // adj2 = zero-diag(oadj) normalized by sqrt row sums
  so[i][j] = (i == j) ? 0.f : ov;
  __syncthreads();
  if (t < 16) {
    float s = 0.f;
    for (int l = 0; l < 16; ++l) s += so[t][l];
    dcol[t] = sqrtf(s) + EPSf;
  }
  __syncthreads();
  adj2[((size_t)g * K2d + i) * K2d + j] = so[i][j] / (dcol[i] * dcol[j]);
}

// ---------------- conv2 part 1: Ah2 = adj2 @ xp ----------------
__global__ void k_conv2a(const float* __restrict__ adj2, const float* __restrict__ xp,
                         float* __restrict__ Ah2) {
  int idx = blockIdx.x * blockDim.x + threadIdx.x;   // B*16*64
  if (idx >= Bg * K2d * Hd) return;
  int g = idx / (K2d * Hd);
  int rem = idx - g * K2d * Hd;
  int i = rem >> 6, fc = rem & 63;
  const float* ar = adj2 + ((size_t)g * K2d + i) * K2d;
  const float* xr = xp + (size_t)g * K2d * Hd + fc;
  float acc = 0.f;
  for (int jj = 0; jj < K2d; ++jj) acc += ar[jj] * xr[(size_t)jj * Hd];
  Ah2[idx] = acc;
}

// ---------------- conv2 part 2: h2c = Ah2@relW + relB + xp@rootW ----------------
__global__ void k_conv2b(const float* __restrict__ Ah2, const float* __restrict__ xp,
                         const float* __restrict__ relw, const float* __restrict__ relb,
                         const float* __restrict__ rootw, float* __restrict__ h2c) {
  int idx = blockIdx.x * blockDim.x + threadIdx.x;
  if (idx >= Bg * K2d * Hd) return;
  int row = idx >> 6, fc = idx & 63;
  const float* ar = Ah2 + (size_t)row * Hd;
  const float* xr = xp + (size_t)row * Hd;
  float acc = relb[fc];
  for (int k = 0; k < Hd; ++k)
    acc += ar[k] * relw[k * Hd + fc] + xr[k] * rootw[k * Hd + fc];
  h2c[idx] = acc;
}

// ---------------- readout + MLP + log_softmax ----------------
__global__ void k_head(const float* __restrict__ h2c, const float* __restrict__ w2,
                       const float* __restrict__ b2, const float* __restrict__ w3,
                       const float* __restrict__ b3, float* __restrict__ outp) {
  __shared__ float r[Hd], hh[Hd], lg[OUTd];
  int g = blockIdx.x, f = threadIdx.x;
  float acc = 0.f;
  for (int i = 0; i < K2d; ++i) acc += h2c[((size_t)g * K2d + i) * Hd + f];
  r[f] = acc;
  __syncthreads();
  float a2 = b2[f];
  for (int k = 0; k < Hd; ++k) a2 += r[k] * w2[k * Hd + f];
  hh[f] = fmaxf(a2, 0.f);
  __syncthreads();
  if (f < OUTd) {
    float a3 = b3[f];
    for (int k = 0; k < Hd; ++k) a3 += hh[k] * w3[k * OUTd + f];
    lg[f] = a3;
  }
  __syncthreads();
  if (f == 0) {
    float m = lg[0];
    for (int o = 1; o < OUTd; ++o) m = fmaxf(m, lg[o]);
    float se = 0.f;
    for (int o = 0; o < OUTd; ++o) se += expf(lg[o] - m);
    float lse = m + logf(se);
    for (int o = 0; o < OUTd; ++o) outp[g * OUTd + o] = lg[o] - lse;
  }
}

// ---------------- scalar finalize (deterministic reduce) ----------------
__global__ void k_finalize(const float* __restrict__ stat, float* __restrict__ outp) {
  __shared__ float sm[Bg], so[Bg];
  int t = threadIdx.x;
  sm[t] = stat[t] + stat[2 * Bg + t];
  so[t] = stat[Bg + t] + stat[3 * Bg + t];
  __syncthreads();
  for (int s = 32; s > 0; s >>= 1) {
    if (t < s) { sm[t] += sm[t + s]; so[t] += so[t + s]; }
    __syncthreads();
  }
  if (t == 0) { outp[Bg * OUTd] = sm[0] / (float)Bg; outp[Bg * OUTd + 1] = so[0] / (float)Bg; }
}

// ---------------- launch ----------------
extern "C" void kernel_launch(void* const* d_in, const int* in_sizes, int n_in,
                              void* d_out, int out_size, void* d_ws, size_t ws_size,
                              hipStream_t stream) {
  const float* x     = (const float*)d_in[0];
  const int*   ei    = (const int*)d_in[1];
  const float* w1    = (const float*)d_in[3];
  const float* b1    = (const float*)d_in[4];
  const float* p1w   = (const float*)d_in[5];
  const float* p1b   = (const float*)d_in[6];
  const float* p2w   = (const float*)d_in[7];
  const float* p2b   = (const float*)d_in[8];
  const float* c1rw  = (const float*)d_in[9];
  const float* c1rb  = (const float*)d_in[10];
  const float* c1tw  = (const float*)d_in[11];
  const float* c2rw  = (const float*)d_in[12];
  const float* c2rb  = (const float*)d_in[13];
  const float* c2tw  = (const float*)d_in[14];
  const float* w2    = (const float*)d_in[15];
  const float* b2    = (const float*)d_in[16];
  const float* w3    = (const float*)d_in[17];
  const float* b3    = (const float*)d_in[18];
  float* outp = (float*)d_out;

  float* ws = (float*)d_ws;
  size_t off = 0;
  float* adj  = ws + off; off += (size_t)Bg * Nn * Nn;
  float* h    = ws + off; off += (size_t)Bg * Nn * Hd;
  float* hn   = ws + off; off += (size_t)Bg * Nn * Hd;
  float* Ah   = ws + off; off += (size_t)Bg * Nn * Hd;
  float* s1   = ws + off; off += (size_t)Bg * Nn * 2;
  float* As1  = ws + off; off += (size_t)Bg * Nn * 2;
  float* deg  = ws + off; off += (size_t)Bg * Nn;
  float* fv   = ws + off; off += (size_t)Bg * Nn;
  float* coef = ws + off; off += (size_t)Bg * 2;
  float* s2   = ws + off; off += (size_t)Bg * Nn * K2d;
  float* As2  = ws + off; off += (size_t)Bg * Nn * K2d;
  float* deg2 = ws + off; off += (size_t)Bg * Nn;
  float* xp   = ws + off; off += (size_t)Bg * K2d * Hd;
  float* oadj = ws + off; off += (size_t)Bg * K2d * K2d;
  float* ssb  = ws + off; off += (size_t)Bg * K2d * K2d;
  float* adj2 = ws + off; off += (size_t)Bg * K2d * K2d;
  float* Ah2  = ws + off; off += (size_t)Bg * K2d * Hd;
  float* h2c  = ws + off; off += (size_t)Bg * K2d * Hd;
  float* stat = ws + off; off += (size_t)4 * Bg;

  hipMemsetAsync(adj, 0, (size_t)Bg * Nn * Nn * sizeof(float), stream);

  k_build_adj<<<Etot / 256, 256, 0, stream>>>(ei, adj);
  k_lin1<<<(Bg * Nn / 16) / 4, 128, 0, stream>>>(x, w1, b1, h);
  k_pool1<<<(Bg * Nn) / 256, 256, 0, stream>>>(h, p1w, p1b, s1);
  k_as1_deg<<<(Bg * Nn) / 256, 256, 0, stream>>>(adj, s1, As1, deg);
  k_stats1<<<Bg, 256, 0, stream>>>(s1, As1, deg, coef, stat);
  k_fvec<<<(Bg * Nn) / 256, 256, 0, stream>>>(s1, coef, fv);
  k_rewire<<<(int)(((size_t)Bg * Nn * Nn / 4) / 256), 256, 0, stream>>>(adj, fv);
  k_spmm<<<(Bg * (Nn / 16)) / 4, 128, 0, stream>>>(adj, h, Ah);
  k_convmix<<<(Bg * Nn / 16) / 4, 128, 0, stream>>>(Ah, h, c1rw, c1rb, c1tw, hn);
  k_pool2<<<(Bg * Nn) / 256, 256, 0, stream>>>(hn, p2w, p2b, s2);
  k_spmm_s2<<<(Bg * (Nn / 16)) / 4, 128, 0, stream>>>(adj, s2, As2);
  k_deg2<<<(Bg * Nn) / 256, 256, 0, stream>>>(adj, deg2);
  k_poolx<<<Bg / 4, 128, 0, stream>>>(s2, hn, xp);
  k_oadj_ss<<<Bg / 4, 128, 0, stream>>>(s2, As2, oadj, ssb);
  k_stats2<<<Bg, 256, 0, stream>>>(s2, deg2, oadj, ssb, adj2, stat);
  k_conv2a<<<(Bg * K2d * Hd) / 256, 256, 0, stream>>>(adj2, xp, Ah2);
  k_conv2b<<<(Bg * K2d * Hd) / 256, 256, 0, stream>>>(Ah2, xp, c2rw, c2rb, c2tw, h2c);
  k_head<<<Bg, Hd, 0, stream>>>(h2c, w2, b2, w3, b3, outp);
  k_finalize<<<1, Bg, 0, stream>>>(stat, outp);
}